// _DualEncoderModule_57363583205828
// MI455X (gfx1250) — compile-verified
//
#include <hip/hip_runtime.h>

#define H 128

typedef __attribute__((ext_vector_type(2))) float    v2f;
typedef __attribute__((ext_vector_type(8))) float    v8f;
typedef __attribute__((ext_vector_type(4))) unsigned v4u;
typedef __attribute__((ext_vector_type(8))) int      v8i;
typedef __attribute__((ext_vector_type(4))) int      v4i;

// ---------------------------------------------------------------- counts
__global__ void count_edges(const int* __restrict__ dst, float* __restrict__ cnt, int E) {
  int e = blockIdx.x * blockDim.x + threadIdx.x;
  if (e < E) atomicAdd(&cnt[dst[e]], 1.0f);
}

// ------------------------------------------------- segment-sum (scatter)
__global__ __launch_bounds__(256) void scatter_add_rows(
    const float* __restrict__ xsrc, const int* __restrict__ src,
    const int* __restrict__ dst, float* __restrict__ agg, int E) {
  int e = blockIdx.x * 8 + (threadIdx.x >> 5);
  if (e >= E) return;
  int lane = threadIdx.x & 31;
  int s = src[e];
  int d = dst[e];
  const float4 v = *(const float4*)(xsrc + (size_t)s * H + lane * 4);
  float* p = agg + (size_t)d * H + lane * 4;
  atomicAdd(p + 0, v.x);
  atomicAdd(p + 1, v.y);
  atomicAdd(p + 2, v.z);
  atomicAdd(p + 3, v.w);
}

// ------------------------------------------- mean scaling (in place)
__global__ __launch_bounds__(256) void scale_rows(
    float* __restrict__ agg, const float* __restrict__ cnt, int n) {
  int i = blockIdx.x * 256 + threadIdx.x;
  int row = i >> 5, lane = i & 31;
  if (row >= n) return;
  float inv = 1.0f / fmaxf(cnt[row], 1.0f);
  float4* p = (float4*)(agg + (size_t)row * H + lane * 4);
  float4 v = *p;
  v.x *= inv; v.y *= inv; v.z *= inv; v.w *= inv;
  *p = v;
}

// ---------------------------------------------------- TDM 2D tile load
// Loads a [tile_rows x 32-float] tile (row stride ld_elems floats) into LDS
// at byte offset lds_off, padding 2 DWORDs every 32 DWORDs -> LDS row
// stride of 34 floats (conflict-free for the WMMA fragment reads below).
__device__ __forceinline__ void tdm_load_2d(unsigned lds_off, const float* gptr,
                                            unsigned ld_elems, unsigned rows_rem,
                                            unsigned tile_rows) {
  unsigned long long ga = (unsigned long long)(uintptr_t)gptr;
  v4u g0;
  g0[0] = 1u;                                   // count=1 (valid user D#)
  g0[1] = lds_off;                              // LDS byte address
  g0[2] = (unsigned)ga;                         // global addr [31:0]
  g0[3] = (unsigned)((ga >> 32) & 0x01FFFFFFu)  // global addr [56:32]
          | (2u << 30);                         // type = 2 ("image")
  v8i g1;
  g1[0] = (int)((2u << 16)      // data_size = 4B
                | (1u << 20)    // pad_enable
                | (4u << 22)    // pad_interval code 4 -> every 32 DWORDs
                | (1u << 25));  // pad_amount  code 1 -> 2 DWORDs
  g1[1] = (int)(32u << 16);                          // tensor_dim0 = 32
  g1[2] = (int)((rows_rem & 0xFFFFu) << 16);         // tensor_dim1 lo16
  g1[3] = (int)((rows_rem >> 16) & 0xFFFFu)          // tensor_dim1 hi16
          | (int)(32u << 16);                        // tile_dim0 = 32
  g1[4] = (int)(tile_rows & 0xFFFFu);                // tile_dim1 (tile_dim2=0)
  g1[5] = (int)ld_elems;                             // tensor_dim0_stride lo32
  g1[6] = 0;
  g1[7] = 0;
  v4i z4 = {0, 0, 0, 0};
  v8i z8 = {0, 0, 0, 0, 0, 0, 0, 0};
  // clang-23 / therock lane: 6-arg form (v4u, v8i, v4i, v4i, v8i, i32 cpol)
  __builtin_amdgcn_tensor_load_to_lds(g0, g1, z4, z4, z8, 0);
}

// ---------------------------------------------------------- WMMA GEMM
// Y[rows, N] = maybe_relu( [X0 | X1] @ [W0 | W1]^T + bias )
//   W0 [N,K0], W1 [N,K1] row-major; N = gridDim.y*128; K0+K1 multiple of 32.
//   blockDim = 128 (4 waves); wave = 16 rows x 128 cols of output.
//   Panels are streamed by the Tensor Data Mover into double-buffered
//   padded LDS; chunk i+1's DMA is in flight while chunk i is consumed
//   (s_wait_tensorcnt 1 = newest may still be outstanding).
__global__ __launch_bounds__(128) void gemm_xwt(
    const float* __restrict__ X0, int ldx0,
    const float* __restrict__ X1, int ldx1, int K0, int K1,
    const float* __restrict__ W0, const float* __restrict__ W1,
    const float* __restrict__ bias,
    float* __restrict__ Y, int ldy, int rows, int relu) {
  __shared__ __attribute__((aligned(16))) float lds_w[2][128 * 34];  // W chunk [n][k]
  __shared__ __attribute__((aligned(16))) float lds_a[2][64 * 34];   // A chunk [row][k]

  const int tid   = threadIdx.x;
  const int wv    = tid >> 5;
  const int lane  = tid & 31;
  const int m     = lane & 15;         // tile row (A) / tile col (B)
  const int kq    = (lane >> 4) << 1;  // half-wave K offset (0 or 2)
  const int rowb0 = blockIdx.x * 64;
  const int colb  = blockIdx.y * 128;
  const int K     = K0 + K1;
  const int nchunk = K >> 5;

  v8f acc[8];
#pragma unroll
  for (int t = 0; t < 8; ++t) {
    float bvv = bias[colb + t * 16 + m];
#pragma unroll
    for (int r = 0; r < 8; ++r) acc[t][r] = bvv;
  }

  const unsigned rows_rem = (unsigned)(rows - rowb0);

  // issue the TDM loads for the chunk starting at K-offset kc into buffer buf
  auto issue_chunk = [&](int kc, int buf) {
    if (wv == 0) {  // W chunk: 128 output cols x 32 k
      const float* wp;
      unsigned ld;
      if (kc < K0) { wp = W0 + (size_t)colb * K0 + kc;        ld = (unsigned)K0; }
      else         { wp = W1 + (size_t)colb * K1 + (kc - K0); ld = (unsigned)K1; }
      tdm_load_2d((unsigned)(uintptr_t)(void*)lds_w[buf], wp, ld, 128u, 128u);
    } else if (wv == 1) {  // A chunk: 64 rows x 32 k
      const float* ap;
      unsigned ld;
      if (kc < K0) { ap = X0 + (size_t)rowb0 * ldx0 + kc;        ld = (unsigned)ldx0; }
      else         { ap = X1 + (size_t)rowb0 * ldx1 + (kc - K0); ld = (unsigned)ldx1; }
      tdm_load_2d((unsigned)(uintptr_t)(void*)lds_a[buf], ap, ld, rows_rem, 64u);
    }
  };

  issue_chunk(0, 0);

  for (int i = 0; i < nchunk; ++i) {
    const int buf = i & 1;
    if (i + 1 < nchunk) {
      // prefetch next chunk into the other buffer (it was last read in
      // iteration i-1, whose closing barrier has already been passed)
      issue_chunk((i + 1) << 5, buf ^ 1);
      if (wv <= 1) __builtin_amdgcn_s_wait_tensorcnt(1);  // chunk i landed
    } else {
      if (wv <= 1) __builtin_amdgcn_s_wait_tensorcnt(0);
    }
    __syncthreads();

    const float* la = lds_a[buf] + (size_t)(wv * 16 + m) * 34;
    const float* lw = lds_w[buf] + (size_t)m * 34;
#pragma unroll
    for (int k0 = 0; k0 < 32; k0 += 4) {
      // A 16x4 f32: lanes 0-15 K={k0,k0+1}, lanes 16-31 K={k0+2,k0+3}
      v2f a = *(const v2f*)(la + k0 + kq);
#pragma unroll
      for (int t = 0; t < 8; ++t) {
        // B 4x16 f32 mirrors A: b = { W[n][k], W[n][k+1] }, adjacent in LDS
        v2f b = *(const v2f*)(lw + (size_t)t * 16 * 34 + k0 + kq);
        acc[t] = __builtin_amdgcn_wmma_f32_16x16x4_f32(
            false, a, false, b, (short)0, acc[t], false, false);
      }
    }
    __syncthreads();
  }

  // C/D layout: VGPR r -> M = r (lanes 0-15) / r+8 (lanes 16-31), N = m
  const int srow0 = rowb0 + wv * 16 + ((lane >> 4) << 3);
#pragma unroll
  for (int r = 0; r < 8; ++r) {
    int row = srow0 + r;
    if (row < rows) {
#pragma unroll
      for (int t = 0; t < 8; ++t) {
        float v = acc[t][r];
        if (relu) v = fmaxf(v, 0.0f);
        Y[(size_t)row * ldy + colb + t * 16 + m] = v;
      }
    }
  }
}

// ------------------------------------------ gather batch rows into LINK
__global__ __launch_bounds__(256) void gather_link(
    const float* __restrict__ dx, const float* __restrict__ sx,
    const int* __restrict__ didx, const int* __restrict__ sidx,
    float* __restrict__ link, int B) {
  int i = blockIdx.x * 256 + threadIdx.x;
  int row = i >> 5, lane = i & 31;
  if (row >= B) return;
  float4 a = *(const float4*)(dx + (size_t)didx[row] * H + lane * 4);
  float4 b = *(const float4*)(sx + (size_t)sidx[row] * H + lane * 4);
  *(float4*)(link + (size_t)row * 4 * H + lane * 4) = a;
  *(float4*)(link + (size_t)row * 4 * H + H + lane * 4) = b;
}

// ------- collapse len-1 MHA:  CA[m] = Wout[m] @ Wv[m],  cb = Wout@bv + bout
__global__ void attn_combine(const float* __restrict__ Win,
                             const float* __restrict__ bin,
                             const float* __restrict__ Wout,
                             const float* __restrict__ bout,
                             float* __restrict__ CA, float* __restrict__ cb) {
  int mm = blockIdx.x >> 7;
  int o  = blockIdx.x & 127;
  int k  = threadIdx.x;
  const float* wout_row = Wout + ((size_t)mm * H + o) * H;
  const float* wvp = Win + (size_t)mm * (3 * H * H) + (size_t)2 * H * H;  // Wv
  float s = 0.0f;
  for (int j = 0; j < H; ++j) s += wout_row[j] * wvp[(size_t)j * H + k];
  CA[((size_t)mm * H + o) * H + k] = s;
  if (k == 0) {
    const float* bvp = bin + (size_t)mm * 3 * H + 2 * H;
    float sb = bout[mm * H + o];
    for (int j = 0; j < H; ++j) sb += wout_row[j] * bvp[j];
    cb[mm * H + o] = sb;
  }
}

// ---------------------------------------------------- final N=1 GEMV
__global__ __launch_bounds__(256) void final_out(
    const float* __restrict__ h2, const float* __restrict__ W3,
    const float* __restrict__ b3, float* __restrict__ out, int B) {
  int i = blockIdx.x * 256 + threadIdx.x;
  int row = i >> 5, lane = i & 31;
  if (row >= B) return;
  float4 hv = *(const float4*)(h2 + (size_t)row * H + lane * 4);
  float4 wvv = *(const float4*)(W3 + lane * 4);
  float v = hv.x * wvv.x + hv.y * wvv.y + hv.z * wvv.z + hv.w * wvv.w;
  for (int off = 16; off > 0; off >>= 1) v += __shfl_xor(v, off, 32);
  if (lane == 0) out[row] = v + b3[0];
}

// ---------------------------------------------------------------- host
static inline void launch_gemm(const float* X0, int ldx0,
                               const float* X1, int ldx1, int K0, int K1,
                               const float* W0, const float* W1,
                               const float* bias, float* Y, int ldy,
                               int rows, int N, int relu, hipStream_t stream) {
  dim3 grid((rows + 63) / 64, N / 128);
  gemm_xwt<<<grid, 128, 0, stream>>>(X0, ldx0, X1, ldx1, K0, K1,
                                     W0, W1, bias, Y, ldy, rows, relu);
}

extern "C" void kernel_launch(void* const* d_in, const int* in_sizes, int n_in,
                              void* d_out, int out_size, void* d_ws, size_t ws_size,
                              hipStream_t stream) {
  const int*   src_d2di  = (const int*)d_in[0];
  const int*   dst_d2di  = (const int*)d_in[1];
  const int*   src_di2dr = (const int*)d_in[2];
  const int*   dst_di2dr = (const int*)d_in[3];
  const int*   drug_idx  = (const int*)d_in[4];
  const int*   dis_idx   = (const int*)d_in[5];
  const float* emb_drug  = (const float*)d_in[6];
  const float* emb_dis   = (const float*)d_in[7];
  const float* conv_Wl   = (const float*)d_in[8];
  const float* conv_bl   = (const float*)d_in[9];
  const float* conv_Wr   = (const float*)d_in[10];
  const float* attn_Win  = (const float*)d_in[11];
  const float* attn_bin  = (const float*)d_in[12];
  const float* attn_Wout = (const float*)d_in[13];
  const float* attn_bout = (const float*)d_in[14];
  const float* mlp_W1    = (const float*)d_in[15];
  const float* mlp_b1    = (const float*)d_in[16];
  const float* mlp_W2    = (const float*)d_in[17];
  const float* mlp_b2    = (const float*)d_in[18];
  const float* mlp_W3    = (const float*)d_in[19];
  const float* mlp_b3    = (const float*)d_in[20];
  float* out = (float*)d_out;

  const int E     = in_sizes[0];
  const int B     = in_sizes[4];
  const int NDRUG = in_sizes[6] / H;
  const int NDIS  = in_sizes[7] / H;
  const int Lnum  = in_sizes[8] / (2 * 2 * H * H);  // == 2

  float* ws = (float*)d_ws;
  size_t o = 0;
  float* cnt_dis  = ws + o; o += (size_t)NDIS;
  float* cnt_drug = ws + o; o += (size_t)NDRUG;
  float* agg_dis  = ws + o; o += (size_t)NDIS * H;
  float* agg_drug = ws + o; o += (size_t)NDRUG * H;
  float* xs_s0    = ws + o; o += (size_t)NDIS * H;   // stack0 dis l0
  float* xd_s0    = ws + o; o += (size_t)NDRUG * H;  // stack0 drug l0 -> drug_x
  float* xs_s1    = ws + o; o += (size_t)NDIS * H;   // stack1 dis l0 -> dis_x
  float* xd_s1    = ws + o; o += (size_t)NDRUG * H;  // stack1 drug l0
  float* link     = ws + o; o += (size_t)B * 4 * H;
  float* h1       = ws + o; o += (size_t)B * 2 * H;
  float* h2       = ws + o; o += (size_t)B * H;
  float* CA       = ws + o; o += (size_t)2 * H * H;
  float* cb       = ws + o; o += (size_t)2 * H;
  (void)ws_size; (void)n_in; (void)out_size;

  auto wl = [&](int s, int l, int e) { return conv_Wl + (size_t)((s * Lnum + l) * 2 + e) * H * H; };
  auto wr = [&](int s, int l, int e) { return conv_Wr + (size_t)((s * Lnum + l) * 2 + e) * H * H; };
  auto bl = [&](int s, int l, int e) { return conv_bl + (size_t)((s * Lnum + l) * 2 + e) * H; };

  // degree counts (shared by everything)
  (void)hipMemsetAsync(cnt_dis, 0, (size_t)(NDIS + NDRUG) * sizeof(float), stream);
  count_edges<<<(E + 255) / 256, 256, 0, stream>>>(dst_d2di, cnt_dis, E);
  count_edges<<<(E + 255) / 256, 256, 0, stream>>>(dst_di2dr, cnt_drug, E);

  // ---- layer 0: aggregations are weight-independent -> shared by both stacks
  (void)hipMemsetAsync(agg_dis, 0, (size_t)NDIS * H * sizeof(float), stream);
  (void)hipMemsetAsync(agg_drug, 0, (size_t)NDRUG * H * sizeof(float), stream);
  scatter_add_rows<<<(E + 7) / 8, 256, 0, stream>>>(emb_drug, src_d2di, dst_d2di, agg_dis, E);
  scatter_add_rows<<<(E + 7) / 8, 256, 0, stream>>>(emb_dis, src_di2dr, dst_di2dr, agg_drug, E);
  scale_rows<<<(NDIS * 32 + 255) / 256, 256, 0, stream>>>(agg_dis, cnt_dis, NDIS);
  scale_rows<<<(NDRUG * 32 + 255) / 256, 256, 0, stream>>>(agg_drug, cnt_drug, NDRUG);

  // layer-0 fused GEMM + bias + relu (per stack)
  launch_gemm(agg_dis, H, emb_dis, H, H, H, wl(0, 0, 0), wr(0, 0, 0), bl(0, 0, 0),
              xs_s0, H, NDIS, H, 1, stream);
  launch_gemm(agg_dis, H, emb_dis, H, H, H, wl(1, 0, 0), wr(1, 0, 0), bl(1, 0, 0),
              xs_s1, H, NDIS, H, 1, stream);
  launch_gemm(agg_drug, H, emb_drug, H, H, H, wl(0, 0, 1), wr(0, 0, 1), bl(0, 0, 1),
              xd_s0, H, NDRUG, H, 1, stream);
  launch_gemm(agg_drug, H, emb_drug, H, H, H, wl(1, 0, 1), wr(1, 0, 1), bl(1, 0, 1),
              xd_s1, H, NDRUG, H, 1, stream);

  // ---- layer 1, stack 0: only drug output needed (edge_type 1, dis->drug)
  (void)hipMemsetAsync(agg_drug, 0, (size_t)NDRUG * H * sizeof(float), stream);
  scatter_add_rows<<<(E + 7) / 8, 256, 0, stream>>>(xs_s0, src_di2dr, dst_di2dr, agg_drug, E);
  scale_rows<<<(NDRUG * 32 + 255) / 256, 256, 0, stream>>>(agg_drug, cnt_drug, NDRUG);
  launch_gemm(agg_drug, H, xd_s0, H, H, H, wl(0, 1, 1), wr(0, 1, 1), bl(0, 1, 1),
              xd_s0 /* in place -> drug_x */, H, NDRUG, H, 1, stream);

  // ---- layer 1, stack 1: only disease output needed (edge_type 0, drug->dis)
  (void)hipMemsetAsync(agg_dis, 0, (size_t)NDIS * H * sizeof(float), stream);
  scatter_add_rows<<<(E + 7) / 8, 256, 0, stream>>>(xd_s1, src_d2di, dst_d2di, agg_dis, E);
  scale_rows<<<(NDIS * 32 + 255) / 256, 256, 0, stream>>>(agg_dis, cnt_dis, NDIS);
  launch_gemm(agg_dis, H, xs_s1, H, H, H, wl(1, 1, 0), wr(1, 1, 0), bl(1, 1, 0),
              xs_s1 /* in place -> dis_x */, H, NDIS, H, 1, stream);

  // ---- tail: gather batch, collapsed attention, MLP
  gather_link<<<(B * 32 + 255) / 256, 256, 0, stream>>>(xd_s0, xs_s1, drug_idx, dis_idx, link, B);
  attn_combine<<<2 * H, H, 0, stream>>>(attn_Win, attn_bin, attn_Wout, attn_bout, CA, cb);

  // drug_att = SE @ CA0^T + cb0 -> link[:,2H:3H]
  launch_gemm(link + H, 4 * H, nullptr, 0, H, 0, CA, nullptr, cb,
              link + 2 * H, 4 * H, B, H, 0, stream);
  // dis_att  = DE @ CA1^T + cb1 -> link[:,3H:4H]
  launch_gemm(link, 4 * H, nullptr, 0, H, 0, CA + H * H, nullptr, cb + H,
              link + 3 * H, 4 * H, B, H, 0, stream);

  // MLP: 4H -> 2H -> H -> 1
  launch_gemm(link, 4 * H, nullptr, 0, 4 * H, 0, mlp_W1, nullptr, mlp_b1,
              h1, 2 * H, B, 2 * H, 1, stream);
  launch_gemm(h1, 2 * H, nullptr, 0, 2 * H, 0, mlp_W2, nullptr, mlp_b2,
              h2, H, B, H, 1, stream);
  final_out<<<(B * 32 + 255) / 256, 256, 0, stream>>>(h2, mlp_W3, mlp_b3, out, B);
}